// ICNN_net_1503238553972
// MI455X (gfx1250) — compile-verified
//
#include <hip/hip_runtime.h>

// ============================================================================
// ICNN Lyapunov-projected dynamics, MI455X (gfx1250, wave32, WMMA).
//
// Roofline: ~83 GFLOP fp32 vs ~8 MB HBM traffic -> compute bound. All
// 64x64 / 128x128 per-sample matmuls are mapped to V_WMMA_F32_16X16X4_F32
// (full fp32 precision on the matrix pipe), 16 samples per wave.
//
// Assumes ws_size >= 294912 bytes (4x 128x128 + 2x 64x64 fp32 transposes).
// N (=500000) must be a multiple of 32 (16 samples/wave * 2 waves/block).
// ============================================================================

#define NW     2          // waves per block
#define WSZ    32
#define ST64   68         // padded LDS row stride (floats) for 64-wide stages
#define ST128  132        // padded LDS row stride (floats) for 128-wide stages

typedef float v2f __attribute__((ext_vector_type(2)));
typedef float v8f __attribute__((ext_vector_type(8)));

__device__ __forceinline__ v8f wmma4(v2f a, v2f b, v8f c) {
  // D[16x16] = A[16x4] * B[4x16] + C   (fp32, matrix core)
  return __builtin_amdgcn_wmma_f32_16x16x4_f32(
      /*neg_a=*/false, a, /*neg_b=*/false, b, /*c_mod=*/(short)0, c,
      /*reuse_a=*/false, /*reuse_b=*/false);
}

__device__ __forceinline__ float srelu(float x) {
  return x < 0.f ? 0.f : (x < 1.f ? 0.5f * x * x : x - 0.5f);
}
__device__ __forceinline__ float sreluD(float x) {
  return x < 0.f ? 0.f : (x < 1.f ? x : 1.f);
}
__device__ __forceinline__ float lrelu(float x) { return x >= 0.f ? x : 0.01f * x; }

// Same-wave LDS producer/consumer ordering: wavefront fence (lowers to the
// needed s_wait_dscnt) + scheduling barrier. No workgroup barrier required,
// every stage buffer is private to one wave.
__device__ __forceinline__ void wsync() {
  __builtin_amdgcn_fence(__ATOMIC_ACQ_REL, "wavefront");
  __builtin_amdgcn_wave_barrier();
}

// ---------------------------------------------------------------------------
// V-net hidden layer forward: dst = srelu(d @ Wx.T + src @ Wz.T), keep mask.
// src/dst/msk: [16][ST64] LDS. BT = Wz^T (k-major, ld=64, coalesced). 
// ---------------------------------------------------------------------------
__device__ __forceinline__ void vlayer_fwd(const float* __restrict__ src,
                                           float* __restrict__ dstS,
                                           float* __restrict__ msk,
                                           const float* __restrict__ BT,
                                           const float* __restrict__ Wx,
                                           const float* __restrict__ dS,
                                           int col, int half) {
  v8f acc[4];
#pragma unroll
  for (int ob = 0; ob < 4; ++ob) {
    int n = ob * 16 + col;
    float w0 = Wx[n * 2 + 0], w1 = Wx[n * 2 + 1];
#pragma unroll
    for (int r = 0; r < 8; ++r) {
      int m = r + 8 * half;                       // C init = d @ Wx.T
      acc[ob][r] = dS[m * 2 + 0] * w0 + dS[m * 2 + 1] * w1;
    }
  }
  int row = col;
#pragma unroll
  for (int kb = 0; kb < 16; ++kb) {
    int k = kb * 4 + 2 * half;                    // A frag: v[j] = A[row][2h+j]
    v2f a; a.x = src[row * ST64 + k]; a.y = src[row * ST64 + k + 1];
#pragma unroll
    for (int ob = 0; ob < 4; ++ob) {
      int n = ob * 16 + col;                      // B frag: v[j] = B[2h+j][n]
      v2f b; b.x = BT[k * 64 + n]; b.y = BT[(k + 1) * 64 + n];
      acc[ob] = wmma4(a, b, acc[ob]);
    }
  }
#pragma unroll
  for (int ob = 0; ob < 4; ++ob) {
    int n = ob * 16 + col;
#pragma unroll
    for (int r = 0; r < 8; ++r) {
      int m = r + 8 * half;
      float aa = acc[ob][r];
      dstS[m * ST64 + n] = srelu(aa);
      msk[m * ST64 + n] = sreluD(aa);
    }
  }
}

// ---------------------------------------------------------------------------
// V-net backward layer: dst = (src @ Wz) * msk.  Bnat = Wz row-major (ld=64).
// ---------------------------------------------------------------------------
__device__ __forceinline__ void vlayer_bwd(const float* __restrict__ src,
                                           float* __restrict__ dstS,
                                           const float* __restrict__ msk,
                                           const float* __restrict__ Bnat,
                                           int col, int half) {
  v8f acc[4];
#pragma unroll
  for (int ob = 0; ob < 4; ++ob)
#pragma unroll
    for (int r = 0; r < 8; ++r) acc[ob][r] = 0.f;
  int row = col;
#pragma unroll
  for (int kb = 0; kb < 16; ++kb) {
    int k = kb * 4 + 2 * half;
    v2f a; a.x = src[row * ST64 + k]; a.y = src[row * ST64 + k + 1];
#pragma unroll
    for (int ob = 0; ob < 4; ++ob) {
      int n = ob * 16 + col;
      v2f b; b.x = Bnat[k * 64 + n]; b.y = Bnat[(k + 1) * 64 + n];
      acc[ob] = wmma4(a, b, acc[ob]);
    }
  }
#pragma unroll
  for (int ob = 0; ob < 4; ++ob) {
    int n = ob * 16 + col;
#pragma unroll
    for (int r = 0; r < 8; ++r) {
      int m = r + 8 * half;
      dstS[m * ST64 + n] = acc[ob][r] * msk[m * ST64 + n];
    }
  }
}

// ---------------------------------------------------------------------------
// f-hat hidden layer: dst = lrelu(src @ W.T + b).  WT = W^T (ld=128).
// ---------------------------------------------------------------------------
__device__ __forceinline__ void flayer(const float* __restrict__ src,
                                       float* __restrict__ dstS,
                                       const float* __restrict__ WT,
                                       const float* __restrict__ bias,
                                       int col, int half) {
  v8f acc[8];
#pragma unroll
  for (int ob = 0; ob < 8; ++ob) {
    float bn = bias[ob * 16 + col];
#pragma unroll
    for (int r = 0; r < 8; ++r) acc[ob][r] = bn;
  }
  int row = col;
#pragma unroll 4
  for (int kb = 0; kb < 32; ++kb) {
    int k = kb * 4 + 2 * half;
    v2f a; a.x = src[row * ST128 + k]; a.y = src[row * ST128 + k + 1];
#pragma unroll
    for (int ob = 0; ob < 8; ++ob) {
      int n = ob * 16 + col;
      v2f b; b.x = WT[k * 128 + n]; b.y = WT[(k + 1) * 128 + n];
      acc[ob] = wmma4(a, b, acc[ob]);
    }
  }
#pragma unroll
  for (int ob = 0; ob < 8; ++ob) {
    int n = ob * 16 + col;
#pragma unroll
    for (int r = 0; r < 8; ++r) {
      int m = r + 8 * half;
      dstS[m * ST128 + n] = lrelu(acc[ob][r]);
    }
  }
}

// ---------------------------------------------------------------------------
__global__ void transpose_k(const float* __restrict__ src, float* __restrict__ dst,
                            int R, int C) {
  int i = blockIdx.x * blockDim.x + threadIdx.x;
  if (i < R * C) {
    int r = i / C, c = i - r * C;
    dst[c * R + r] = src[i];
  }
}

// ---------------------------------------------------------------------------
__global__ __launch_bounds__(NW * WSZ) void icnn_kernel(
    const float* __restrict__ X, const float* __restrict__ Xs,
    const float* __restrict__ Vl1, const float* __restrict__ V2x,
    const float* __restrict__ V2z, const float* __restrict__ V3x,
    const float* __restrict__ V3z, const float* __restrict__ Vfx,
    const float* __restrict__ Vfz, const float* __restrict__ f1w,
    const float* __restrict__ f1b, const float* __restrict__ f2b,
    const float* __restrict__ f3b, const float* __restrict__ f4b,
    const float* __restrict__ f5b, const float* __restrict__ ffw,
    const float* __restrict__ ffb, const float* __restrict__ WT2,
    const float* __restrict__ WT3, const float* __restrict__ WT4,
    const float* __restrict__ WT5, const float* __restrict__ V2zT,
    const float* __restrict__ V3zT, float* __restrict__ out, int nTiles) {
  __shared__ float stA[NW][16 * ST128];   // ping activation stage
  __shared__ float stB[NW][16 * ST128];   // pong activation stage
  __shared__ float mk1[NW][16 * ST64];    // srelu' masks for backward
  __shared__ float mk2[NW][16 * ST64];
  __shared__ float mk3[NW][16 * ST64];
  __shared__ float dstg[NW][32];          // d = X - Xstable, [16][2]

  const int wv = threadIdx.x >> 5;
  const int lane = threadIdx.x & 31;
  const int col = lane & 15;
  const int half = lane >> 4;
  const int tile = blockIdx.x * NW + wv;
  if (tile >= nTiles) return;             // wave-uniform

  float* sA = stA[wv];
  float* sB = stB[wv];
  float* M1 = mk1[wv];
  float* M2 = mk2[wv];
  float* M3 = mk3[wv];
  float* dS = dstg[wv];

  const int gb = tile * 16;               // first sample of this tile

  // ---- load d = X - Xstable (32 contiguous floats = 16 samples x 2) -------
  dS[lane] = X[gb * 2 + lane] - Xs[gb * 2 + lane];
  wsync();
  const float d0 = dS[col * 2 + 0];       // this lane's sample = col
  const float d1 = dS[col * 2 + 1];

  // ============================ V forward ==================================
  // z1 = srelu(d @ Vl1.T)   (K=2: plain VALU, store z1 + mask)
#pragma unroll
  for (int ob = 0; ob < 4; ++ob) {
    int n = ob * 16 + col;
    float w0 = Vl1[n * 2 + 0], w1 = Vl1[n * 2 + 1];
#pragma unroll
    for (int r = 0; r < 8; ++r) {
      int m = r + 8 * half;
      float a = dS[m * 2 + 0] * w0 + dS[m * 2 + 1] * w1;
      sA[m * ST64 + n] = srelu(a);
      M1[m * ST64 + n] = sreluD(a);
    }
  }
  wsync();
  vlayer_fwd(sA, sB, M2, V2zT, V2x, dS, col, half);   // z2
  wsync();
  vlayer_fwd(sB, sA, M3, V3zT, V3x, dS, col, half);   // z3 -> sA
  wsync();

  // af = d @ Vfx.T + z3 @ Vfz.T  (scalar per sample; halves duplicate)
  float af = d0 * Vfx[0] + d1 * Vfx[1];
#pragma unroll 8
  for (int k = 0; k < 64; ++k) af += sA[col * ST64 + k] * Vfz[k];
  const float zf = srelu(af);
  const float Vval = srelu(zf) + 0.01f * (d0 * d0 + d1 * d1);   // TOL=0.01
  const float gf = sreluD(zf) * sreluD(af);
  wsync();

  // ============================ V backward =================================
  // g3 = gf * Vfz * srelu'(a3)  -> sB   (halves split the 64 features)
#pragma unroll 8
  for (int kk = 0; kk < 32; ++kk) {
    int k = half * 32 + kk;
    sB[col * ST64 + k] = gf * Vfz[k] * M3[col * ST64 + k];
  }
  wsync();

  float dd0 = gf * Vfx[0] + 0.02f * d0;   // 2*TOL*d + gf*Vfx
  float dd1 = gf * Vfx[1] + 0.02f * d1;
#pragma unroll 8
  for (int k = 0; k < 64; ++k) {          // dd += g3 @ V3x
    float g = sB[col * ST64 + k];
    dd0 += g * V3x[k * 2 + 0];
    dd1 += g * V3x[k * 2 + 1];
  }
  vlayer_bwd(sB, sA, M2, V3z, col, half); // g2 = (g3 @ V3z) * m2 -> sA
  wsync();
#pragma unroll 8
  for (int k = 0; k < 64; ++k) {          // dd += g2 @ V2x
    float g = sA[col * ST64 + k];
    dd0 += g * V2x[k * 2 + 0];
    dd1 += g * V2x[k * 2 + 1];
  }
  vlayer_bwd(sA, sB, M1, V2z, col, half); // g1 = (g2 @ V2z) * m1 -> sB
  wsync();
#pragma unroll 8
  for (int k = 0; k < 64; ++k) {          // dd += g1 @ Vl1
    float g = sB[col * ST64 + k];
    dd0 += g * Vl1[k * 2 + 0];
    dd1 += g * Vl1[k * 2 + 1];
  }
  wsync();

  // ============================ f-hat MLP ==================================
  // h1 = lrelu(d @ f1w.T + f1b)  (K=2: VALU) -> sA (128-wide)
#pragma unroll
  for (int ob = 0; ob < 8; ++ob) {
    int n = ob * 16 + col;
    float w0 = f1w[n * 2 + 0], w1 = f1w[n * 2 + 1], bn = f1b[n];
#pragma unroll
    for (int r = 0; r < 8; ++r) {
      int m = r + 8 * half;
      sA[m * ST128 + n] = lrelu(dS[m * 2 + 0] * w0 + dS[m * 2 + 1] * w1 + bn);
    }
  }
  wsync();
  flayer(sA, sB, WT2, f2b, col, half); wsync();
  flayer(sB, sA, WT3, f3b, col, half); wsync();
  flayer(sA, sB, WT4, f4b, col, half); wsync();
  flayer(sB, sA, WT5, f5b, col, half); wsync();

  // ff: lane(half=c) computes component c of f_hat for sample col
  float fh = ffb[half];
#pragma unroll 8
  for (int k = 0; k < 128; ++k) fh += sA[col * ST128 + k] * ffw[half * 128 + k];

  // =========================== final combine ===============================
  const float fho = __shfl_xor(fh, 16, 32);
  const float fh0 = (half == 0) ? fh : fho;
  const float fh1 = (half == 0) ? fho : fh;
  const float Vnorm = dd0 * dd0 + dd1 * dd1;
  const float fhV = fh0 * dd0 + fh1 * dd1;
  float num = fhV + 0.1f * Vval;                       // ALPHA=0.1
  num = num > 0.f ? num : 0.f;
  const float fmul = num / (Vnorm + 1e-10f);           // EPS
  const float mydd = (half == 0) ? dd0 : dd1;
  out[(gb + col) * 2 + half] = fh - mydd * fmul;
}

// ===========================================================================
extern "C" void kernel_launch(void* const* d_in, const int* in_sizes, int n_in,
                              void* d_out, int out_size, void* d_ws,
                              size_t ws_size, hipStream_t stream) {
  const float* X   = (const float*)d_in[0];
  const float* Xs  = (const float*)d_in[1];
  const float* Vl1 = (const float*)d_in[2];
  const float* V2x = (const float*)d_in[3];
  const float* V2z = (const float*)d_in[4];
  const float* V3x = (const float*)d_in[5];
  const float* V3z = (const float*)d_in[6];
  const float* Vfx = (const float*)d_in[7];
  const float* Vfz = (const float*)d_in[8];
  const float* f1w = (const float*)d_in[9];
  const float* f1b = (const float*)d_in[10];
  const float* f2w = (const float*)d_in[11];
  const float* f2b = (const float*)d_in[12];
  const float* f3w = (const float*)d_in[13];
  const float* f3b = (const float*)d_in[14];
  const float* f4w = (const float*)d_in[15];
  const float* f4b = (const float*)d_in[16];
  const float* f5w = (const float*)d_in[17];
  const float* f5b = (const float*)d_in[18];
  const float* ffw = (const float*)d_in[19];
  const float* ffb = (const float*)d_in[20];

  float* ws = (float*)d_ws;                 // needs >= 73728 floats
  float* WT2  = ws + 0;
  float* WT3  = ws + 16384;
  float* WT4  = ws + 32768;
  float* WT5  = ws + 49152;
  float* V2zT = ws + 65536;
  float* V3zT = ws + 69632;

  const int T = 256;
  transpose_k<<<(16384 + T - 1) / T, T, 0, stream>>>(f2w, WT2, 128, 128);
  transpose_k<<<(16384 + T - 1) / T, T, 0, stream>>>(f3w, WT3, 128, 128);
  transpose_k<<<(16384 + T - 1) / T, T, 0, stream>>>(f4w, WT4, 128, 128);
  transpose_k<<<(16384 + T - 1) / T, T, 0, stream>>>(f5w, WT5, 128, 128);
  transpose_k<<<(4096 + T - 1) / T, T, 0, stream>>>(V2z, V2zT, 64, 64);
  transpose_k<<<(4096 + T - 1) / T, T, 0, stream>>>(V3z, V3zT, 64, 64);

  const int N = in_sizes[0] / 2;            // 500000
  const int nTiles = N / 16;                // 31250, exact
  const int nBlocks = nTiles / NW;          // 15625, exact

  icnn_kernel<<<nBlocks, NW * WSZ, 0, stream>>>(
      X, Xs, Vl1, V2x, V2z, V3x, V3z, Vfx, Vfz, f1w, f1b, f2b, f3b, f4b, f5b,
      ffw, ffb, WT2, WT3, WT4, WT5, V2zT, V3zT, (float*)d_out, nTiles);
}